// EmbedLayer0AttentionWrapper_49478023250151
// MI455X (gfx1250) — compile-verified
//
#include <hip/hip_runtime.h>
#include <cstdint>

// ---- model constants (match reference) ----
#define H      1536
#define NH     12
#define NKV    2
#define HD     128
#define CTX    4096
#define NLAYERS 28
#define EPS    1e-6f
#define ROPE_BASE 1000000.0f

typedef float v2f __attribute__((ext_vector_type(2)));
typedef float v8f __attribute__((ext_vector_type(8)));

static __device__ __forceinline__ v8f wmma_f32(v2f a, v2f b, v8f c) {
  return __builtin_amdgcn_wmma_f32_16x16x4_f32(
      /*neg_a=*/false, a, /*neg_b=*/false, b,
      /*c_mod=*/(short)0, c, /*reuse_a=*/false, /*reuse_b=*/false);
}

// ---------------------------------------------------------------------------
// Kernel 1: embedding gather + RMSNorm.  One block, 256 threads.
// ---------------------------------------------------------------------------
__global__ void embed_norm_kernel(const int* __restrict__ ids,
                                  const float* __restrict__ embed_w,
                                  const float* __restrict__ ln_w,
                                  float* __restrict__ h,
                                  float* __restrict__ n) {
  __shared__ float red[256];
  const int tid = threadIdx.x;
  const int row = ids[0];
  const float4* e4 = (const float4*)(embed_w + (size_t)row * H);
  float ss = 0.f;
  for (int j = tid; j < H / 4; j += 256) {   // 384 float4 elements
    float4 v = e4[j];
    ((float4*)h)[j] = v;
    ss += v.x * v.x + v.y * v.y + v.z * v.z + v.w * v.w;
  }
  red[tid] = ss;
  __syncthreads();
  for (int s = 128; s > 0; s >>= 1) {
    if (tid < s) red[tid] += red[tid + s];
    __syncthreads();
  }
  const float rms = rsqrtf(red[0] * (1.0f / (float)H) + EPS);
  for (int j = tid; j < H / 4; j += 256) {
    float4 v = ((const float4*)h)[j];
    float4 g = ((const float4*)ln_w)[j];
    float4 o;
    o.x = v.x * rms * g.x; o.y = v.y * rms * g.y;
    o.z = v.z * rms * g.z; o.w = v.w * rms * g.w;
    ((float4*)n)[j] = o;
  }
}

// ---------------------------------------------------------------------------
// Kernel 2: GEMV via V_WMMA_F32_16X16X4_F32, cols fixed to H.
//   y[r] = sum_i W[r,i]*x[i] (+bias)(+res), W row-major [rows x H].
//
// K-chunk permutation trick: a GEMV may contract columns in any order, so
// lane-half hh takes columns {i+4hh .. i+4hh+3} from ONE b128 load and feeds
// TWO wmmas ((x,y) then (z,w)), with matching b128 LDS broadcasts for B.
// Two independent accumulator chains keep the matrix pipe busy.
//
// Layout (per ISA 7.12.2, 32-bit A 16x4):
//   A: lane L (0..15) = row M=L holds {K0,K1}; lane L+16 holds {K2,K3}
//   B: VGPR0 = {K0 | K2}, VGPR1 = {K1 | K3} by lane-half
//   D: VGPR v, lane-half hh -> M = v + 8*hh (all 16 N columns identical)
// Each wave produces 16 output rows; 8 waves/block; grid.x = rows/128.
// x is DMA-staged into LDS with global_load_async_to_lds_b128 (ASYNCcnt).
// ---------------------------------------------------------------------------
__global__ void gemv_wmma_kernel(const float* __restrict__ W,
                                 const float* __restrict__ x,
                                 const float* __restrict__ bias,
                                 const float* __restrict__ res,
                                 float* __restrict__ y) {
  __shared__ float sx[H];
  const int tid = threadIdx.x;

  // --- async DMA: global x -> LDS (16B per lane per issue) ---
  for (int i = tid; i < H / 4; i += 256) {
    uint32_t lds_addr = (uint32_t)(uintptr_t)(&sx[i << 2]);
    uint32_t goff     = (uint32_t)(i << 4);
    asm volatile("global_load_async_to_lds_b128 %0, %1, %2"
                 :: "v"(lds_addr), "v"(goff), "s"(x)
                 : "memory");
  }
  asm volatile("s_wait_asynccnt 0" ::: "memory");
  __syncthreads();

  const int lane = tid & 31;
  const int wave = tid >> 5;
  const int row0 = (blockIdx.x * 8 + wave) * 16;
  const int m    = lane & 15;
  const int hh   = lane >> 4;  // lane half

  // lane-half hh owns a 4-column sub-chunk at offset 4*hh
  const float* wrow = W + (size_t)(row0 + m) * H + (hh << 2);
  const float* sxh  = sx + (hh << 2);

  v8f acc0 = {0.f, 0.f, 0.f, 0.f, 0.f, 0.f, 0.f, 0.f};
  v8f acc1 = {0.f, 0.f, 0.f, 0.f, 0.f, 0.f, 0.f, 0.f};

  #pragma unroll 2
  for (int i = 0; i < H; i += 16) {          // 16 columns / iteration
    float4 w0 = *(const float4*)(wrow + i);        // cols i+4hh   .. +3
    float4 w1 = *(const float4*)(wrow + i + 8);    // cols i+8+4hh .. +3
    float4 x0 = *(const float4*)(sxh + i);         // matching broadcasts
    float4 x1 = *(const float4*)(sxh + i + 8);
    acc0 = wmma_f32((v2f){w0.x, w0.y}, (v2f){x0.x, x0.y}, acc0);
    acc1 = wmma_f32((v2f){w0.z, w0.w}, (v2f){x0.z, x0.w}, acc1);
    acc0 = wmma_f32((v2f){w1.x, w1.y}, (v2f){x1.x, x1.y}, acc0);
    acc1 = wmma_f32((v2f){w1.z, w1.w}, (v2f){x1.z, x1.w}, acc1);
  }
  v8f acc = acc0 + acc1;

  if (m == 0) {  // lanes 0 and 16 hold rows row0+{0..7} and row0+{8..15}
    #pragma unroll
    for (int v = 0; v < 8; ++v) {
      const int r = row0 + v + (hh << 3);
      float val = acc[v];
      if (bias) val += bias[r];
      if (res)  val += res[r];
      y[r] = val;
    }
  }
}

// ---------------------------------------------------------------------------
// Kernel 3: RoPE in place on q (12 heads) and k_new (2 heads).
// One thread per rotation pair: 14 heads * 64 pairs = 896 threads.
// ---------------------------------------------------------------------------
__global__ void rope_kernel(float* __restrict__ q, float* __restrict__ k,
                            const int* __restrict__ posp) {
  const int tid  = threadIdx.x;
  const int head = tid >> 6;
  const int j    = tid & 63;
  if (head >= NH + NKV) return;
  const float pos  = (float)posp[0];
  const float invf = powf(ROPE_BASE, -(float)j * (1.0f / 64.0f));
  const float ang  = pos * invf;
  float s, c;
  sincosf(ang, &s, &c);
  float* base = (head < NH) ? (q + head * HD) : (k + (head - NH) * HD);
  const float x1 = base[j];
  const float x2 = base[j + 64];
  base[j]      = x1 * c - x2 * s;
  base[j + 64] = x2 * c + x1 * s;
}

// ---------------------------------------------------------------------------
// Kernel 4: attention for one head per block (12 blocks, 256 threads).
// Reads cache rows for t < pos and the new (RoPE'd) k/v for t == pos,
// so the input kv_cache is never mutated. Mask == truncation at pos.
// ---------------------------------------------------------------------------
__global__ void attn_kernel(const float* __restrict__ q,
                            const float* __restrict__ kn,
                            const float* __restrict__ vn,
                            const float* __restrict__ cache,
                            const int* __restrict__ posp,
                            float* __restrict__ ctx) {
  __shared__ float sc[CTX];
  __shared__ float red[256];
  __shared__ float part[2][HD];

  const int hidx = blockIdx.x;
  const int tid  = threadIdx.x;
  const int kv   = hidx / (NH / NKV);
  const int pos  = posp[0];

  const float* qh = q + hidx * HD;
  const float* Kc = cache + (size_t)kv * CTX * HD;                     // kv_cache[0][kv]
  const float* Vc = cache + ((size_t)NLAYERS * NKV + kv) * CTX * HD;   // kv_cache[NLAYERS][kv]
  const float scale = 0.08838834764831845f;  // 1/sqrt(128)

  // scores
  for (int t = tid; t <= pos; t += 256) {
    const float* kr = (t == pos) ? (kn + kv * HD) : (Kc + (size_t)t * HD);
    const float4* k4 = (const float4*)kr;
    const float4* q4 = (const float4*)qh;
    float acc = 0.f;
    #pragma unroll 8
    for (int d = 0; d < HD / 4; ++d) {
      float4 a = q4[d], b = k4[d];
      acc += a.x * b.x + a.y * b.y + a.z * b.z + a.w * b.w;
    }
    sc[t] = acc * scale;
  }
  __syncthreads();

  // softmax: max
  float mx = -1e30f;
  for (int t = tid; t <= pos; t += 256) mx = fmaxf(mx, sc[t]);
  red[tid] = mx;
  __syncthreads();
  for (int s = 128; s > 0; s >>= 1) {
    if (tid < s) red[tid] = fmaxf(red[tid], red[tid + s]);
    __syncthreads();
  }
  mx = red[0];
  __syncthreads();

  // softmax: exp + sum
  float sum = 0.f;
  for (int t = tid; t <= pos; t += 256) {
    float e = expf(sc[t] - mx);
    sc[t] = e;
    sum += e;
  }
  red[tid] = sum;
  __syncthreads();
  for (int s = 128; s > 0; s >>= 1) {
    if (tid < s) red[tid] += red[tid + s];
    __syncthreads();
  }
  const float inv = 1.0f / red[0];

  // context: 2 t-slices x 128 dims (coalesced across lanes in d)
  const int d    = tid & (HD - 1);
  const int half = tid >> 7;
  float acc = 0.f;
  for (int t = half; t <= pos; t += 2) {
    const float* vr = (t == pos) ? (vn + kv * HD) : (Vc + (size_t)t * HD);
    acc += sc[t] * vr[d];
  }
  part[half][d] = acc;
  __syncthreads();
  if (half == 0) ctx[hidx * HD + d] = (part[0][d] + part[1][d]) * inv;
}

// ---------------------------------------------------------------------------
// Host side
// ---------------------------------------------------------------------------
extern "C" void kernel_launch(void* const* d_in, const int* in_sizes, int n_in,
                              void* d_out, int out_size, void* d_ws, size_t ws_size,
                              hipStream_t stream) {
  const int*   ids     = (const int*)d_in[0];
  // d_in[1] position_ids, d_in[2] causal_mask: unused (pos truncation is exact)
  const int*   posp    = (const int*)d_in[3];
  const float* embed_w = (const float*)d_in[4];
  const float* ln_w    = (const float*)d_in[5];
  const float* q_w     = (const float*)d_in[6];
  const float* q_b     = (const float*)d_in[7];
  const float* k_w     = (const float*)d_in[8];
  const float* k_b     = (const float*)d_in[9];
  const float* v_w     = (const float*)d_in[10];
  const float* v_b     = (const float*)d_in[11];
  const float* o_w     = (const float*)d_in[12];
  const float* cache   = (const float*)d_in[13];
  float*       out     = (float*)d_out;

  float* ws   = (float*)d_ws;
  float* h    = ws;            // [1536]
  float* nrm  = ws + H;        // [1536]
  float* qb   = ws + 2 * H;    // [1536]
  float* kb   = ws + 3 * H;    // [256]
  float* vb   = ws + 3 * H + NKV * HD;          // [256]
  float* ctxb = ws + 3 * H + 2 * NKV * HD;      // [1536]

  embed_norm_kernel<<<1, 256, 0, stream>>>(ids, embed_w, ln_w, h, nrm);

  // QKV projections (rows/128 blocks, 8 waves * 16 rows each)
  gemv_wmma_kernel<<<(NH * HD) / 128, 256, 0, stream>>>(q_w, nrm, q_b, nullptr, qb);
  gemv_wmma_kernel<<<(NKV * HD) / 128, 256, 0, stream>>>(k_w, nrm, k_b, nullptr, kb);
  gemv_wmma_kernel<<<(NKV * HD) / 128, 256, 0, stream>>>(v_w, nrm, v_b, nullptr, vb);

  rope_kernel<<<1, (NH + NKV) * 64, 0, stream>>>(qb, kb, posp);

  attn_kernel<<<NH, 256, 0, stream>>>(qb, kb, vb, cache, posp, ctxb);

  // output projection + residual (h) straight into d_out
  gemv_wmma_kernel<<<H / 128, 256, 0, stream>>>(o_w, ctxb, nullptr, h, out);
}